// AGCNBlock_51599737094694
// MI455X (gfx1250) — compile-verified
//
#include <hip/hip_runtime.h>

// ---------------- problem constants ----------------
#define Nn   64
#define Cc   64
#define Tt   300
#define Vv   25
#define TV   7500        // T*V
#define ICn  16
#define KTk  9
#define EPSf 1e-5f

typedef float v2f __attribute__((ext_vector_type(2)));
typedef float v8f __attribute__((ext_vector_type(8)));

// D = A(16x4) * B(4x16) + C  -- fp32 WMMA, full fp32 semantics
static __device__ __forceinline__ v8f wmma4(v2f a, v2f b, v8f c) {
    return __builtin_amdgcn_wmma_f32_16x16x4_f32(false, a, false, b, (short)0, c,
                                                 false, false);
}

// Cooperative 2D tile fill: dst[r][c] (row stride = width) =
//   srcRowBase[r*TV + startElem + c], zero outside [0, TV).
static __device__ __forceinline__ void coop_fill(float* dst, const float* srcRowBase,
                                                 int startElem, int width, int tid) {
    for (int idx = tid; idx < Cc * width; idx += 256) {
        const int r = idx / width, c = idx % width;
        const int e = startElem + c;
        dst[idx] = (e >= 0 && e < TV) ? srcRowBase[(size_t)r * TV + e] : 0.0f;
    }
}

// ---------------- Tensor Data Mover (CDNA5 TDM) ----------------
#if __has_builtin(__builtin_amdgcn_tensor_load_to_lds)
#define USE_TDM 1
typedef unsigned int u32x4 __attribute__((ext_vector_type(4)));
typedef int          i32x4 __attribute__((ext_vector_type(4)));
typedef int          i32x8 __attribute__((ext_vector_type(8)));

// DMA a 2D tile [nrows x width] (f32, row stride TV elems) from global to LDS.
// Row elements at x >= rowRemain read as zero (hardware OOB) -> right padding.
// Descriptor bit layout per CDNA5 ISA.
static __device__ __forceinline__ void tdm_load_2d(float* ldsDst, const float* gSrc,
                                                   int rowRemain, int width, int nrows) {
    const unsigned long long ga = (unsigned long long)(uintptr_t)gSrc;
    const unsigned lds = (unsigned)(uintptr_t)ldsDst;  // addr[31:0] = LDS byte offset
    const unsigned td0 = (unsigned)rowRemain;          // tensor_dim0 (OOB -> zero)
    // group0: count=1 | lds_addr | global_addr[56:0] | type=2 (bits 127:126)
    u32x4 g0 = { 1u, lds, (unsigned)ga, (unsigned)(ga >> 32) | 0x80000000u };
    // group1: data_size=4B; tensor_dim0=td0; tensor_dim1=nrows; tile_dim0=width;
    //         tile_dim1=nrows; tensor_dim0_stride=TV; dim1 stride unused (2D tile)
    i32x8 g1 = { (int)0x00020000u,
                 (int)((td0 & 0xFFFFu) << 16),
                 (int)((td0 >> 16) | ((unsigned)nrows << 16)),
                 (int)(((unsigned)width) << 16),
                 nrows,
                 TV, 0, 0 };
    i32x4 gz = { 0, 0, 0, 0 };
#if __clang_major__ >= 23
    // amdgpu-toolchain / clang-23: 6-arg form (extra int32x8 before cpol)
    i32x8 gz8 = { 0, 0, 0, 0, 0, 0, 0, 0 };
    __builtin_amdgcn_tensor_load_to_lds(g0, g1, gz, gz, gz8, 0);
#else
    // ROCm 7.2 / clang-22: 5-arg form
    __builtin_amdgcn_tensor_load_to_lds(g0, g1, gz, gz, 0);
#endif
    __builtin_amdgcn_s_wait_tensorcnt(0);
}
#endif

// =====================================================================
// K1: fa = Wa[k] @ x[n]  (16x64)@(64x7500), fb likewise (shared B frag)
// grid (118, 64, 3), block 128 (4 waves, 1 16x16 tile each)
// =====================================================================
__global__ void k_fafb(const float* __restrict__ x, const float* __restrict__ Wa,
                       const float* __restrict__ ba, const float* __restrict__ Wb,
                       const float* __restrict__ bb, float* __restrict__ fa,
                       float* __restrict__ fb) {
    const int lane = threadIdx.x & 31;
    const int wid  = threadIdx.x >> 5;
    const int hi = lane >> 4, l15 = lane & 15;
    const int tile = blockIdx.x * 4 + wid;
    const int n = blockIdx.y, k = blockIdx.z;
    const int col  = tile * 16 + l15;
    const int colc = (col < TV) ? col : (TV - 1);

    const float* xp  = x + (size_t)n * Cc * TV + colc;
    const float* WaK = Wa + (size_t)(k * ICn + l15) * Cc;  // row m = l15
    const float* WbK = Wb + (size_t)(k * ICn + l15) * Cc;

    v8f accA = {}; v8f accB = {};
    for (int cc = 0; cc < Cc; cc += 4) {
        const int kb = cc + 2 * hi;
        v2f aA, aB, b;
        aA.x = WaK[kb];  aA.y = WaK[kb + 1];
        aB.x = WbK[kb];  aB.y = WbK[kb + 1];
        b.x = xp[(size_t)kb * TV];  b.y = xp[(size_t)(kb + 1) * TV];
        accA = wmma4(aA, b, accA);
        accB = wmma4(aB, b, accB);
    }
    if (col < TV) {
        float* faP = fa + (size_t)(k * Nn + n) * ICn * TV + col;
        float* fbP = fb + (size_t)(k * Nn + n) * ICn * TV + col;
#pragma unroll
        for (int i = 0; i < 8; ++i) {
            const int mm = i + 8 * hi;
            faP[(size_t)mm * TV] = accA[i] + ba[k * ICn + mm];
            fbP[(size_t)mm * TV] = accB[i] + bb[k * ICn + mm];
        }
    }
}

// =====================================================================
// K2: S[k,n,v,w] = softmax_v( (A1@A2)/4800 ) + adj + PA
//     fa[k,n] flat is exactly [d=4800][v=25]; A1[v,d]=fa[d,v], A2[d,w]=fb[d,w]
// grid (64, 3), block 256
// =====================================================================
#define DCH 64
__global__ void k_S(const float* __restrict__ fa, const float* __restrict__ fb,
                    const float* __restrict__ adj, const float* __restrict__ PA,
                    float* __restrict__ S) {
    __shared__ float sfa[DCH * Vv];
    __shared__ float sfb[DCH * Vv];
    __shared__ float sS[Vv * Vv];
    const int tid = threadIdx.x;
    const int n = blockIdx.x, k = blockIdx.y;
    const float* faB = fa + (size_t)(k * Nn + n) * ICn * TV;
    const float* fbB = fb + (size_t)(k * Nn + n) * ICn * TV;

    const int p0 = tid, p1 = tid + 256, p2 = tid + 512;
    const int v0 = p0 / Vv, w0 = p0 % Vv;
    const int v1 = p1 / Vv, w1 = p1 % Vv;
    const int v2 = p2 / Vv, w2 = p2 % Vv;
    float acc0 = 0.f, acc1 = 0.f, acc2 = 0.f;

    for (int d0 = 0; d0 < ICn * Tt; d0 += DCH) {
        for (int i = tid; i < DCH * Vv; i += 256) {
            sfa[i] = faB[(size_t)d0 * Vv + i];
            sfb[i] = fbB[(size_t)d0 * Vv + i];
        }
        __syncthreads();
        for (int dd = 0; dd < DCH; ++dd) {
            const float* ra = &sfa[dd * Vv];
            const float* rb = &sfb[dd * Vv];
            if (p0 < Vv * Vv) acc0 += ra[v0] * rb[w0];
            if (p1 < Vv * Vv) acc1 += ra[v1] * rb[w1];
            if (p2 < Vv * Vv) acc2 += ra[v2] * rb[w2];
        }
        __syncthreads();
    }
    const float sc = 1.0f / (float)(ICn * Tt);
    if (p0 < Vv * Vv) sS[p0] = acc0 * sc;
    if (p1 < Vv * Vv) sS[p1] = acc1 * sc;
    if (p2 < Vv * Vv) sS[p2] = acc2 * sc;
    __syncthreads();
    if (tid < Vv) {  // softmax over v (axis -2), per column w
        const int w = tid;
        float mx = -1e30f;
        for (int v = 0; v < Vv; ++v) mx = fmaxf(mx, sS[v * Vv + w]);
        float sum = 0.f;
        for (int v = 0; v < Vv; ++v) sum += __expf(sS[v * Vv + w] - mx);
        const float inv = 1.0f / sum;
        float* So = S + (size_t)(k * Nn + n) * Vv * Vv;
        for (int v = 0; v < Vv; ++v) {
            const float e = __expf(sS[v * Vv + w] - mx) * inv;
            So[v * Vv + w] = e + adj[(k * Vv + v) * Vv + w] + PA[(k * Vv + v) * Vv + w];
        }
    }
}

// =====================================================================
// K3: g[n,:,t,:] = sum_k Wd[k] @ (x_t @ S[k,n]) + sum_k bd[k]
//     per block: n and an 8-t tile (200 cols) staged in LDS via TDM.
// grid (38, 64), block 256 (8 waves), dynamic LDS 105984 B
// =====================================================================
__global__ void k_gcn(const float* __restrict__ x, const float* __restrict__ Wd,
                      const float* __restrict__ bd, const float* __restrict__ S,
                      float* __restrict__ g) {
    extern __shared__ float smem[];
    float* xs  = smem;            // [64][200]
    float* hs  = smem + 12800;    // [64][200]
    float* sSk = smem + 25600;    // [28][32]
    const int tid = threadIdx.x;
    const int lane = tid & 31, wid = tid >> 5;
    const int hi = lane >> 4, l15 = lane & 15;
    const int n = blockIdx.y;
    const int t0 = blockIdx.x * 8;
    const int tcount = (Tt - t0 < 8) ? (Tt - t0) : 8;   // last block: 4
    const int start = t0 * Vv;
    const v8f zero = {};

    const float* xn = x + (size_t)n * Cc * TV;
#ifdef USE_TDM
    if (wid == 0) tdm_load_2d(xs, xn + start, TV - start, 200, Cc);
#else
    coop_fill(xs, xn, start, 200, tid);
#endif

    v8f yacc[4][2];
#pragma unroll
    for (int mt = 0; mt < 4; ++mt)
#pragma unroll
        for (int nt = 0; nt < 2; ++nt) yacc[mt][nt] = zero;

    for (int k = 0; k < 3; ++k) {
        __syncthreads();  // xs ready / previous-iter hs+sSk consumed
        const float* Sk = S + (size_t)(k * Nn + n) * Vv * Vv;
        for (int idx = tid; idx < 28 * 32; idx += 256) {
            const int v = idx >> 5, w = idx & 31;
            sSk[idx] = (v < Vv && w < Vv) ? Sk[v * Vv + w] : 0.0f;
        }
        __syncthreads();

        // GEMM1: hs = Wd[k] (64x64) @ xs (64x200); 4 x 13 tiles over 8 waves
        const float* WdK = Wd + (size_t)k * Cc * Cc;
        for (int tt = wid; tt < 52; tt += 8) {
            const int mt = tt & 3, nt = tt >> 2;
            const int row = mt * 16 + l15;
            const int col = nt * 16 + l15;
            const int colc = (col < 200) ? col : 199;
            v8f acc = zero;
            for (int cc = 0; cc < Cc; cc += 4) {
                const int kb = cc + 2 * hi;
                v2f a, b;
                a.x = WdK[row * Cc + kb];     a.y = WdK[row * Cc + kb + 1];
                b.x = xs[kb * 200 + colc];    b.y = xs[(kb + 1) * 200 + colc];
                acc = wmma4(a, b, acc);
            }
            if (col < 200) {
#pragma unroll
                for (int i = 0; i < 8; ++i)
                    hs[(mt * 16 + i + 8 * hi) * 200 + col] = acc[i];
            }
        }
        __syncthreads();

        // GEMM2: wave wid owns t = wid: yacc += H_t(64x25) @ S(25x25)
        if (wid < tcount) {
            const int t = wid;
#pragma unroll
            for (int mt = 0; mt < 4; ++mt) {
                const int row = mt * 16 + l15;
#pragma unroll
                for (int nt = 0; nt < 2; ++nt) {
                    v8f acc = yacc[mt][nt];
                    for (int vv = 0; vv < 28; vv += 4) {
                        const int kb = vv + 2 * hi;
                        const int kb0 = (kb < Vv) ? kb : (Vv - 1);       // B row zero there
                        const int kb1 = (kb + 1 < Vv) ? kb + 1 : (Vv - 1);
                        v2f a, b;
                        a.x = hs[row * 200 + t * Vv + kb0];
                        a.y = hs[row * 200 + t * Vv + kb1];
                        b.x = sSk[kb * 32 + nt * 16 + l15];
                        b.y = sSk[(kb + 1) * 32 + nt * 16 + l15];
                        acc = wmma4(a, b, acc);
                    }
                    yacc[mt][nt] = acc;
                }
            }
        }
    }

    if (wid < tcount) {
        const int t = t0 + wid;
#pragma unroll
        for (int mt = 0; mt < 4; ++mt)
#pragma unroll
            for (int nt = 0; nt < 2; ++nt)
#pragma unroll
                for (int i = 0; i < 8; ++i) {
                    const int o = mt * 16 + i + 8 * hi;
                    const int w = nt * 16 + l15;
                    if (w < Vv) {
                        const float bsum = bd[o] + bd[64 + o] + bd[128 + o];
                        g[((size_t)(n * Cc + o) * Tt + t) * Vv + w] =
                            yacc[mt][nt][i] + bsum;
                    }
                }
    }
}

// =====================================================================
// K4: per-channel BN stats (deterministic tree reduction). grid 64, block 256
// =====================================================================
__global__ void k_bnstats(const float* __restrict__ in, float* __restrict__ mu,
                          float* __restrict__ istd) {
    __shared__ float ssum[256], ssq[256];
    const int c = blockIdx.x, tid = threadIdx.x;
    float s = 0.f, q = 0.f;
    for (int idx = tid; idx < Nn * TV; idx += 256) {
        const int ni = idx / TV, tvi = idx % TV;
        const float v = in[((size_t)ni * Cc + c) * TV + tvi];
        s += v; q += v * v;
    }
    ssum[tid] = s; ssq[tid] = q;
    __syncthreads();
    for (int off = 128; off > 0; off >>= 1) {
        if (tid < off) { ssum[tid] += ssum[tid + off]; ssq[tid] += ssq[tid + off]; }
        __syncthreads();
    }
    if (tid == 0) {
        const float m = ssum[0] * (1.0f / (float)(Nn * TV));
        const float var = ssq[0] * (1.0f / (float)(Nn * TV)) - m * m;
        mu[c] = m;
        istd[c] = rsqrtf(var + EPSf);
    }
}

// =====================================================================
// K5/K8: out = relu( (in - mu)*istd*gamma + beta + x ), float4. grid 30000x256
// =====================================================================
__global__ void k_bnres(const float* __restrict__ in, const float* __restrict__ x,
                        const float* __restrict__ mu, const float* __restrict__ istd,
                        const float* __restrict__ gamma, const float* __restrict__ beta,
                        float* __restrict__ out) {
    const size_t i4 = (size_t)blockIdx.x * blockDim.x + threadIdx.x;
    if (i4 >= (size_t)Nn * Cc * TV / 4) return;
    const int c = (int)(((i4 * 4) / TV) % Cc);   // TV % 4 == 0 -> c uniform in float4
    const float4 a  = ((const float4*)in)[i4];
    const float4 xr = ((const float4*)x)[i4];
    const float m = mu[c], is = istd[c] * gamma[c], b = beta[c];
    float4 r;
    r.x = fmaxf(0.0f, (a.x - m) * is + b + xr.x);
    r.y = fmaxf(0.0f, (a.y - m) * is + b + xr.y);
    r.z = fmaxf(0.0f, (a.z - m) * is + b + xr.z);
    r.w = fmaxf(0.0f, (a.w - m) * is + b + xr.w);
    ((float4*)out)[i4] = r;
}

// =====================================================================
// K6: temporal conv  y2 = Wt[64x576] @ window(r)[576x200] + bt
//     8-t tile + 4-halo each side staged in LDS ([64][400]) via TDM.
//     K enumerated (dt outer, i inner) so each 4-wide K chunk has fixed dt.
// grid (38, 64), block 256, dynamic LDS 102400 B
// =====================================================================
__global__ void k_tcn(const float* __restrict__ r, const float* __restrict__ Wt,
                      const float* __restrict__ bt, float* __restrict__ y2) {
    extern __shared__ float smem[];
    float* rt = smem;  // [64][400] = [c][(8+8)*25]
    const int tid = threadIdx.x;
    const int lane = tid & 31, wid = tid >> 5;
    const int hi = lane >> 4, l15 = lane & 15;
    const int n = blockIdx.y;
    const int t0 = blockIdx.x * 8;
    const int tcount = (Tt - t0 < 8) ? (Tt - t0) : 8;
    const int ncols = tcount * Vv;
    const int start = (t0 - 4) * Vv;        // window begins 4 t's early
    const float* rn = r + (size_t)n * Cc * TV;

#ifdef USE_TDM
    if (blockIdx.x > 0) {                   // interior: non-negative start,
        if (wid == 0)                       // right padding via hardware OOB
            tdm_load_2d(rt, rn + start, TV - start, 400, Cc);
    } else {
        coop_fill(rt, rn, start, 400, tid); // left halo needs zero fill
    }
#else
    coop_fill(rt, rn, start, 400, tid);
#endif
    __syncthreads();

    for (int tt = wid; tt < 52; tt += 8) {  // 4 x 13 tiles of 16x16
        const int mt = tt & 3, nt = tt >> 2;
        const int row = mt * 16 + l15;
        const int col = nt * 16 + l15;
        const int colc = (col < 200) ? col : 199;
        const int tlc = colc / Vv, v = colc % Vv;
        const float* WtR = Wt + (size_t)row * (Cc * KTk);
        v8f acc = {};
        for (int dt = 0; dt < KTk; ++dt) {
            const int bofs = (tlc + dt) * Vv + v;
            for (int ib = 0; ib < Cc; ib += 4) {
                const int i0 = ib + 2 * hi;
                v2f a, b;
                a.x = WtR[i0 * KTk + dt];
                a.y = WtR[(i0 + 1) * KTk + dt];
                b.x = rt[i0 * 400 + bofs];
                b.y = rt[(i0 + 1) * 400 + bofs];
                acc = wmma4(a, b, acc);
            }
        }
        if (col < ncols) {
#pragma unroll
            for (int i = 0; i < 8; ++i) {
                const int o = mt * 16 + i + 8 * hi;
                y2[((size_t)(n * Cc + o) * Tt + (t0 + tlc)) * Vv + v] = acc[i] + bt[o];
            }
        }
    }
}

__global__ void k_copy(const float* __restrict__ src, float* __restrict__ dst, int n) {
    const int i = blockIdx.x * blockDim.x + threadIdx.x;
    if (i < n) dst[i] = src[i];
}

// =====================================================================
// launch: all on `stream`; ws laid out with lifetime overlap (~247 MB peak)
// =====================================================================
extern "C" void kernel_launch(void* const* d_in, const int* in_sizes, int n_in,
                              void* d_out, int out_size, void* d_ws, size_t ws_size,
                              hipStream_t stream) {
    const float* x   = (const float*)d_in[0];
    const float* adj = (const float*)d_in[1];
    const float* PA  = (const float*)d_in[2];
    const float* Wa  = (const float*)d_in[3];
    const float* ba  = (const float*)d_in[4];
    const float* Wb  = (const float*)d_in[5];
    const float* bb  = (const float*)d_in[6];
    const float* Wd  = (const float*)d_in[7];
    const float* bd  = (const float*)d_in[8];
    const float* g1  = (const float*)d_in[9];
    const float* b1  = (const float*)d_in[10];
    const float* Wt  = (const float*)d_in[11];
    const float* bt  = (const float*)d_in[12];
    const float* g2  = (const float*)d_in[13];
    const float* b2  = (const float*)d_in[14];

    float* ws = (float*)d_ws;
    float* fa = ws;                 // [0, 23.04M)   dead after k_S
    float* fb = ws + 23040000;      // [23.04M, 46.08M) dead after k_S
    float* g  = ws;                 // [0, 30.72M)   reuses fa (+fb head)
    float* rr = ws + 30720000;      // [30.72M, 61.44M) reuses fb tail
    float* S  = ws + 61440000;      // 120000
    float* st = ws + 61560000;      // mu1|istd1|mu2|istd2 (4 x 64)
    float* y2 = ws;                 // reuses g after it is dead
    float* out = (float*)d_out;

    k_fafb   <<<dim3(118, Nn, 3), 128, 0,      stream>>>(x, Wa, ba, Wb, bb, fa, fb);
    k_S      <<<dim3(Nn, 3),      256, 0,      stream>>>(fa, fb, adj, PA, S);
    k_gcn    <<<dim3(38, Nn),     256, 105984, stream>>>(x, Wd, bd, S, g);
    k_bnstats<<<Cc,               256, 0,      stream>>>(g, st, st + 64);
    k_bnres  <<<30000,            256, 0,      stream>>>(g, x, st, st + 64, g1, b1, rr);
    k_tcn    <<<dim3(38, Nn),     256, 102400, stream>>>(rr, Wt, bt, y2);
    k_bnstats<<<Cc,               256, 0,      stream>>>(y2, st + 128, st + 192);
    k_bnres  <<<30000,            256, 0,      stream>>>(y2, x, st + 128, st + 192, g2, b2, out);
    k_copy   <<<8,                256, 0,      stream>>>(adj, out + (size_t)Nn * Cc * TV, 3 * Vv * Vv);
}